// EdgeFeatures_40321152975476
// MI455X (gfx1250) — compile-verified
//
#include <hip/hip_runtime.h>

// Problem constants (match reference)
#define Bc 4
#define Nn 5000
#define Kk 20
#define Hh 128
#define Ee (Nn * Kk)          // 100000 edges per batch
#define WSTRIDE 132           // padded LDS row stride (floats) -> conflict-free b64 reads

typedef float v2f __attribute__((ext_vector_type(2)));
typedef float v8f __attribute__((ext_vector_type(8)));

__device__ __forceinline__ v8f wmma_f32(v2f a, v2f b, v8f c) {
  // D = A(16x4) x B(4x16) + C, fp32, wave32
  return __builtin_amdgcn_wmma_f32_16x16x4_f32(
      /*neg_a=*/false, a, /*neg_b=*/false, b,
      /*c_mod=*/(short)0, c, /*reuse_a=*/false, /*reuse_b=*/false);
}

// ---------------------------------------------------------------------------
// Kernel 1: node projections  Vx_from = x@Vf_w^T + Vf_b,  Vx_to = x@Vt_w^T + Vt_b
// Rows are flat [B*N, H]. One wave computes one 16-row x 128-col tile.
// ---------------------------------------------------------------------------
__global__ __launch_bounds__(128) void node_proj_f32wmma(
    const float* __restrict__ x,
    const float* __restrict__ Vf_w, const float* __restrict__ Vf_b,
    const float* __restrict__ Vt_w, const float* __restrict__ Vt_b,
    float* __restrict__ vx_from, float* __restrict__ vx_to, int ntiles) {
  __shared__ float sF[Hh * WSTRIDE];
  __shared__ float sT[Hh * WSTRIDE];
  __shared__ float sFb[Hh];
  __shared__ float sTb[Hh];

  const int t = threadIdx.x;
  // Cooperative weight load into padded LDS (row-major [o][h], stride WSTRIDE)
  {
    const int col4 = (t & 31) * 4;
    for (int r = (t >> 5); r < Hh; r += 4) {
      float4 wf = *reinterpret_cast<const float4*>(Vf_w + r * Hh + col4);
      float4 wt = *reinterpret_cast<const float4*>(Vt_w + r * Hh + col4);
      *reinterpret_cast<float4*>(&sF[r * WSTRIDE + col4]) = wf;
      *reinterpret_cast<float4*>(&sT[r * WSTRIDE + col4]) = wt;
    }
    if (t < Hh) { sFb[t] = Vf_b[t]; sTb[t] = Vt_b[t]; }
  }
  __syncthreads();

  const int lane = t & 31;
  const int wid  = t >> 5;
  const int n  = lane & 15;   // B/D column within 16-wide tile
  const int hi = lane >> 4;   // half-wave select (k-pair / row+8)

  const int wave  = blockIdx.x * 4 + wid;
  const int nwave = gridDim.x * 4;

  for (int tile = wave; tile < ntiles; tile += nwave) {
    const int r0 = tile * 16;
    const float* pa = x + (size_t)(r0 + n) * Hh + 2 * hi;  // A row m = n

    for (int w = 0; w < 2; ++w) {
      const float* sW = w ? sT : sF;
      const float* sb = w ? sTb : sFb;
      float* outp     = w ? vx_to : vx_from;

      v8f acc[8];
#pragma unroll
      for (int i = 0; i < 8; ++i) acc[i] = (v8f){0.f,0.f,0.f,0.f,0.f,0.f,0.f,0.f};

      for (int kk = 0; kk < 32; ++kk) {
        v2f a = *reinterpret_cast<const v2f*>(pa + 4 * kk);
#pragma unroll
        for (int nt = 0; nt < 8; ++nt) {
          v2f bf = *reinterpret_cast<const v2f*>(
              &sW[(nt * 16 + n) * WSTRIDE + 4 * kk + 2 * hi]);
          acc[nt] = wmma_f32(a, bf, acc[nt]);
        }
      }
#pragma unroll
      for (int vg = 0; vg < 8; ++vg) {
        const int row = r0 + vg + 8 * hi;
        float* po = outp + (size_t)row * Hh;
#pragma unroll
        for (int nt = 0; nt < 8; ++nt) {
          const int o = nt * 16 + n;
          po[o] = acc[nt][vg] + sb[o];
        }
      }
    }
  }
}

// ---------------------------------------------------------------------------
// Kernel 2 (fused): out[b,j,:] = U*e[b,j] + U_b
//                              + (inv==E ? W_ph : iU*e[b,inv] + iU_b)
//                              + Vx_to[b, edge_index[b,j]] + Vx_from[b, j/K]
// Both GEMMs accumulate into one 16x128 fp32 WMMA tile per wave.
// ---------------------------------------------------------------------------
__global__ __launch_bounds__(128) void edge_fused_f32wmma(
    const float* __restrict__ e,
    const int* __restrict__ edge_index,
    const int* __restrict__ inv_index,
    const float* __restrict__ U_w, const float* __restrict__ U_b,
    const float* __restrict__ iU_w, const float* __restrict__ iU_b,
    const float* __restrict__ W_ph,
    const float* __restrict__ vx_from, const float* __restrict__ vx_to,
    float* __restrict__ out) {
  __shared__ float sU[Hh * WSTRIDE];
  __shared__ float sI[Hh * WSTRIDE];
  __shared__ float sUb[Hh];
  __shared__ float sIb[Hh];
  __shared__ float sPh[Hh];

  const int t = threadIdx.x;
  {
    const int col4 = (t & 31) * 4;
    for (int r = (t >> 5); r < Hh; r += 4) {
      float4 wu = *reinterpret_cast<const float4*>(U_w + r * Hh + col4);
      float4 wi = *reinterpret_cast<const float4*>(iU_w + r * Hh + col4);
      *reinterpret_cast<float4*>(&sU[r * WSTRIDE + col4]) = wu;
      *reinterpret_cast<float4*>(&sI[r * WSTRIDE + col4]) = wi;
    }
    if (t < Hh) { sUb[t] = U_b[t]; sIb[t] = iU_b[t]; sPh[t] = W_ph[t]; }
  }
  __syncthreads();

  const int lane = t & 31;
  const int wid  = t >> 5;
  const int n  = lane & 15;
  const int hi = lane >> 4;

  const int tiles_per_b = Ee / 16;          // 6250
  const int total_tiles = Bc * tiles_per_b; // 25000
  const int wave  = blockIdx.x * 4 + wid;
  const int nwave = gridDim.x * 4;

  for (int tid = wave; tid < total_tiles; tid += nwave) {
    const int b  = tid / tiles_per_b;
    const int j0 = (tid - b * tiles_per_b) * 16;
    const size_t ebase = (size_t)b * Ee;

    // Per-lane A rows: direct row (m = n) and gathered row for the inv term.
    const int mrow = j0 + n;
    const float* paU = e + (ebase + (size_t)mrow) * Hh + 2 * hi;
    const int inv = inv_index[ebase + mrow];
    const bool ph = (inv >= Ee);                       // placeholder edge
    const float* paI = e + (ebase + (size_t)(ph ? 0 : inv)) * Hh + 2 * hi;

    v8f acc[8];
#pragma unroll
    for (int i = 0; i < 8; ++i) acc[i] = (v8f){0.f,0.f,0.f,0.f,0.f,0.f,0.f,0.f};

    for (int kk = 0; kk < 32; ++kk) {
      v2f aU = *reinterpret_cast<const v2f*>(paU + 4 * kk);
      v2f a0 = *reinterpret_cast<const v2f*>(paI + 4 * kk);
      v2f aI;
      aI.x = ph ? 0.f : a0.x;   // zero rows for placeholder -> GEMM adds 0
      aI.y = ph ? 0.f : a0.y;
#pragma unroll
      for (int nt = 0; nt < 8; ++nt) {
        v2f bU = *reinterpret_cast<const v2f*>(
            &sU[(nt * 16 + n) * WSTRIDE + 4 * kk + 2 * hi]);
        v2f bI = *reinterpret_cast<const v2f*>(
            &sI[(nt * 16 + n) * WSTRIDE + 4 * kk + 2 * hi]);
        acc[nt] = wmma_f32(aU, bU, acc[nt]);
        acc[nt] = wmma_f32(aI, bI, acc[nt]);
      }
    }

    // Epilogue: biases, placeholder row, Vx gathers, store.
#pragma unroll
    for (int vg = 0; vg < 8; ++vg) {
      const int row = j0 + vg + 8 * hi;
      const int ei   = edge_index[ebase + row];
      const bool phr = (inv_index[ebase + row] >= Ee);
      const int node = row / Kk;   // reshape(b,n,k,h) broadcast source
      const float* pt = vx_to   + ((size_t)b * Nn + ei)   * Hh;
      const float* pf = vx_from + ((size_t)b * Nn + node) * Hh;
      float* po = out + (ebase + (size_t)row) * Hh;
#pragma unroll
      for (int nt = 0; nt < 8; ++nt) {
        const int o = nt * 16 + n;
        const float add = sUb[o] + (phr ? sPh[o] : sIb[o]) + pt[o] + pf[o];
        po[o] = acc[nt][vg] + add;
      }
    }
  }
}

// ---------------------------------------------------------------------------
// Launch
// ---------------------------------------------------------------------------
extern "C" void kernel_launch(void* const* d_in, const int* in_sizes, int n_in,
                              void* d_out, int out_size, void* d_ws, size_t ws_size,
                              hipStream_t stream) {
  const float* x     = (const float*)d_in[0];
  const float* e     = (const float*)d_in[1];
  const int*   eidx  = (const int*)d_in[2];
  const int*   iidx  = (const int*)d_in[3];
  const float* U_w   = (const float*)d_in[4];
  const float* U_b   = (const float*)d_in[5];
  const float* Vf_w  = (const float*)d_in[6];
  const float* Vf_b  = (const float*)d_in[7];
  const float* Vt_w  = (const float*)d_in[8];
  const float* Vt_b  = (const float*)d_in[9];
  const float* iU_w  = (const float*)d_in[10];
  const float* iU_b  = (const float*)d_in[11];
  const float* W_ph  = (const float*)d_in[12];
  float* out = (float*)d_out;

  float* vx_from = (float*)d_ws;                       // [B*N, H]
  float* vx_to   = vx_from + (size_t)Bc * Nn * Hh;     // [B*N, H]

  const int node_tiles = (Bc * Nn) / 16;  // 1250
  node_proj_f32wmma<<<320, 128, 0, stream>>>(x, Vf_w, Vf_b, Vt_w, Vt_b,
                                             vx_from, vx_to, node_tiles);

  edge_fused_f32wmma<<<512, 128, 0, stream>>>(e, eidx, iidx, U_w, U_b,
                                              iU_w, iU_b, W_ph,
                                              vx_from, vx_to, out);
}